// SavedGCN_42425686950472
// MI455X (gfx1250) — compile-verified
//
#include <hip/hip_runtime.h>

// ---------------------------------------------------------------------------
// GCN forward on gfx1250 (MI455X).
//  - Dense transforms use V_WMMA_F32_16X16X4_F32 (fp32 matrix pipe, wave32).
//  - Edge aggregation = edge-parallel global_atomic_add_f32 (L2-resident:
//    h is 25.6MB << 192MB L2, so gathers/atomics live in L2).
//  - Graph mean-pool reduced per-block in LDS, flushed with global atomics.
// ---------------------------------------------------------------------------

typedef float v2f __attribute__((ext_vector_type(2)));
typedef float v8f __attribute__((ext_vector_type(8)));

#define F_DIM 64
#define NGRAPH 64

// deg[i] = 1 (self loop) ; zero pooling accumulators
__global__ void k_init(float* __restrict__ deg, float* __restrict__ sums,
                       float* __restrict__ cnts, int N) {
  int i = blockIdx.x * blockDim.x + threadIdx.x;
  if (i < N) deg[i] = 1.0f;
  if (i < NGRAPH * F_DIM) sums[i] = 0.0f;
  if (i < NGRAPH) cnts[i] = 0.0f;
}

__global__ void k_deg(const int* __restrict__ dst, float* __restrict__ deg, int E) {
  int e = blockIdx.x * blockDim.x + threadIdx.x;
  if (e < E) atomicAdd(&deg[dst[e]], 1.0f);
}

// deg -> dinv in place (deg >= 1 always, so rsqrt is safe)
__global__ void k_rsqrt(float* __restrict__ deg, int N) {
  int i = blockIdx.x * blockDim.x + threadIdx.x;
  if (i < N) deg[i] = rsqrtf(deg[i]);
}

// Y[N,64] = (relu_in ? relu(X) : X) @ W   with W [64,64] row-major.
// One wave -> 16 rows x 64 cols via 4 accumulators; 16 K-steps of WMMA f32 K=4.
__global__ void k_gemm64(const float* __restrict__ X, const float* __restrict__ W,
                         float* __restrict__ Y, int N, int relu_in) {
  const int lane = threadIdx.x & 31;
  const int wave = threadIdx.x >> 5;       // 8 waves / block
  const int hi   = lane >> 4;              // 0: K pair {0,1} / rows r..  1: {2,3} / rows r+8
  const int lo   = lane & 15;
  const int tileRow = (blockIdx.x * 8 + wave) * 16;

  int arow = tileRow + lo;
  if (arow > N - 1) arow = N - 1;          // clamp: keeps EXEC all-ones for WMMA
  const float* __restrict__ Ap = X + (size_t)arow * F_DIM + 2 * hi;

  v8f c0 = {}, c1 = {}, c2 = {}, c3 = {};
#pragma unroll
  for (int k0 = 0; k0 < F_DIM; k0 += 4) {
    v2f a;
    a.x = Ap[k0];
    a.y = Ap[k0 + 1];
    if (relu_in) { a.x = fmaxf(a.x, 0.0f); a.y = fmaxf(a.y, 0.0f); }

    const float* __restrict__ B0 = W + (size_t)(k0 + 2 * hi) * F_DIM + lo;
    v2f b;
    b.x = B0[0];            b.y = B0[F_DIM];
    c0 = __builtin_amdgcn_wmma_f32_16x16x4_f32(false, a, false, b, (short)0, c0, false, false);
    b.x = B0[16];           b.y = B0[F_DIM + 16];
    c1 = __builtin_amdgcn_wmma_f32_16x16x4_f32(false, a, false, b, (short)0, c1, false, false);
    b.x = B0[32];           b.y = B0[F_DIM + 32];
    c2 = __builtin_amdgcn_wmma_f32_16x16x4_f32(false, a, false, b, (short)0, c2, false, false);
    b.x = B0[48];           b.y = B0[F_DIM + 48];
    c3 = __builtin_amdgcn_wmma_f32_16x16x4_f32(false, a, false, b, (short)0, c3, false, false);
  }

  // C/D layout: VGPR r holds rows (tileRow+r) [lanes 0-15] and (tileRow+r+8) [lanes 16-31]
#pragma unroll
  for (int r = 0; r < 8; ++r) {
    int row = tileRow + r + hi * 8;
    if (row < N) {
      float* yr = Y + (size_t)row * F_DIM + lo;
      yr[0]  = c0[r];
      yr[16] = c1[r];
      yr[32] = c2[r];
      yr[48] = c3[r];
    }
  }
}

// AGG[i,f] = HT[i,f]*dinv[i]^2 + b[f]   (self-loop message + bias)
__global__ void k_init_agg(const float* __restrict__ HT, const float* __restrict__ dinv,
                           const float* __restrict__ b, float* __restrict__ AGG, int N) {
  int idx = blockIdx.x * blockDim.x + threadIdx.x;
  if (idx < N * F_DIM) {
    int i = idx >> 6;
    int f = idx & 63;
    float d = dinv[i];
    AGG[idx] = HT[idx] * d * d + b[f];
  }
}

// 16 lanes per edge, 4 features per lane: AGG[dst] += HT[src]*dinv[src]*dinv[dst]
__global__ void k_scatter(const int* __restrict__ src, const int* __restrict__ dst,
                          const float* __restrict__ dinv, const float* __restrict__ HT,
                          float* __restrict__ AGG, int E) {
  int t = blockIdx.x * blockDim.x + threadIdx.x;
  int e = t >> 4;
  if (e >= E) return;
  int c = (t & 15) * 4;
  int j = src[e];
  int i = dst[e];
  float norm = dinv[j] * dinv[i];
  const float4 v = *(const float4*)(HT + (size_t)j * F_DIM + c);
  float* o = AGG + (size_t)i * F_DIM + c;
  atomicAdd(o + 0, v.x * norm);
  atomicAdd(o + 1, v.y * norm);
  atomicAdd(o + 2, v.z * norm);
  atomicAdd(o + 3, v.w * norm);
}

// Per-block LDS reduction into [64 graphs x 64 feats] bins, then atomic flush.
__global__ void k_pool(const float* __restrict__ H, const int* __restrict__ batch,
                       float* __restrict__ sums, float* __restrict__ cnts, int N) {
  __shared__ float sS[NGRAPH * F_DIM];
  __shared__ float sC[NGRAPH];
  int t = threadIdx.x;                       // 256 threads
  for (int i = t; i < NGRAPH * F_DIM; i += 256) sS[i] = 0.0f;
  if (t < NGRAPH) sC[t] = 0.0f;
  __syncthreads();

  int f  = t & 63;
  int nl = t >> 6;                           // 0..3 node lanes
  int base = blockIdx.x * 256;
  for (int i = nl; i < 256; i += 4) {
    int node = base + i;
    if (node < N) {
      int g = batch[node];
      atomicAdd(&sS[g * F_DIM + f], H[(size_t)node * F_DIM + f]);
      if (f == 0) atomicAdd(&sC[g], 1.0f);
    }
  }
  __syncthreads();

  for (int i = t; i < NGRAPH * F_DIM; i += 256) {
    float v = sS[i];
    if (v != 0.0f) atomicAdd(&sums[i], v);   // skipping zeros is a no-op-safe opt
  }
  if (t < NGRAPH) {
    float v = sC[t];
    if (v != 0.0f) atomicAdd(&cnts[t], v);
  }
}

// out[g,o] = blin[o] + sum_f (sums[g,f]/max(cnt,1)) * Wlin[f,o]   (64x2 output)
__global__ void k_final(const float* __restrict__ sums, const float* __restrict__ cnts,
                        const float* __restrict__ Wlin, const float* __restrict__ blin,
                        float* __restrict__ out) {
  int t = threadIdx.x;                       // 128 threads
  int g = t >> 1;
  int o = t & 1;
  float inv = 1.0f / fmaxf(cnts[g], 1.0f);
  float acc = blin[o];
#pragma unroll
  for (int f = 0; f < F_DIM; ++f)
    acc += sums[g * F_DIM + f] * inv * Wlin[f * 2 + o];
  out[t] = acc;
}

extern "C" void kernel_launch(void* const* d_in, const int* in_sizes, int n_in,
                              void* d_out, int out_size, void* d_ws, size_t ws_size,
                              hipStream_t stream) {
  const float* x     = (const float*)d_in[0];
  const int*   ei    = (const int*)d_in[1];
  const int*   batch = (const int*)d_in[2];
  const float* W1    = (const float*)d_in[3];
  const float* b1    = (const float*)d_in[4];
  const float* W2    = (const float*)d_in[5];
  const float* b2    = (const float*)d_in[6];
  const float* W3    = (const float*)d_in[7];
  const float* b3    = (const float*)d_in[8];
  const float* Wlin  = (const float*)d_in[9];
  const float* blin  = (const float*)d_in[10];
  float* out = (float*)d_out;

  const int N = in_sizes[0] / F_DIM;
  const int E = in_sizes[1] / 2;
  const int* src = ei;       // edge_index[0] = message source
  const int* dst = ei + E;   // edge_index[1] = aggregation target

  // workspace layout (floats): dinv[N padded to 64] | HT[N*64] | AGG[N*64] | sums[4096] | cnts[64]
  float* ws   = (float*)d_ws;
  size_t npad = (size_t)((N + 63) & ~63);
  float* dinv = ws;
  float* HT   = dinv + npad;
  float* AGG  = HT + (size_t)N * F_DIM;
  float* sums = AGG + (size_t)N * F_DIM;
  float* cnts = sums + NGRAPH * F_DIM;

  const int B = 256;
  int initN = N > NGRAPH * F_DIM ? N : NGRAPH * F_DIM;

  // degree / normalization
  k_init  <<<(initN + B - 1) / B, B, 0, stream>>>(dinv, sums, cnts, N);
  k_deg   <<<(E + B - 1) / B,     B, 0, stream>>>(dst, dinv, E);
  k_rsqrt <<<(N + B - 1) / B,     B, 0, stream>>>(dinv, N);

  int gemmGrid = (N + 127) / 128;            // 8 waves x 16 rows per block
  int nfGrid   = (N * F_DIM + B - 1) / B;
  int scGrid   = (E * 16 + B - 1) / B;

  // layer 1
  k_gemm64  <<<gemmGrid, B, 0, stream>>>(x, W1, HT, N, 0);
  k_init_agg<<<nfGrid,   B, 0, stream>>>(HT, dinv, b1, AGG, N);
  k_scatter <<<scGrid,   B, 0, stream>>>(src, dst, dinv, HT, AGG, E);
  // layer 2 (relu of layer-1 output fused into GEMM A-load)
  k_gemm64  <<<gemmGrid, B, 0, stream>>>(AGG, W2, HT, N, 1);
  k_init_agg<<<nfGrid,   B, 0, stream>>>(HT, dinv, b2, AGG, N);
  k_scatter <<<scGrid,   B, 0, stream>>>(src, dst, dinv, HT, AGG, E);
  // layer 3 (no relu on output)
  k_gemm64  <<<gemmGrid, B, 0, stream>>>(AGG, W3, HT, N, 1);
  k_init_agg<<<nfGrid,   B, 0, stream>>>(HT, dinv, b3, AGG, N);
  k_scatter <<<scGrid,   B, 0, stream>>>(src, dst, dinv, HT, AGG, E);

  // mean pool + linear head
  k_pool <<<(N + 255) / 256, 256, 0, stream>>>(AGG, batch, sums, cnts, N);
  k_final<<<1, 128, 0, stream>>>(sums, cnts, Wlin, blin, out);
}